// YoloLoss_5660766896341
// MI455X (gfx1250) — compile-verified
//
#include <hip/hip_runtime.h>
#include <hip/hip_bf16.h>
#include <cstdint>
#include <math.h>

#define IOU_THR 0.5f
#define PEN_FAC 0.1f
#define N_PRED 8192
#define N_GT   512
#define BLK    256
#define BLOCKS_PER_BATCH (N_PRED / BLK)   // 32
#define NBATCH 16
#define NWAVES (BLK / 32)                 // 8

// ---------------------------------------------------------------------------
// Main kernel: one block = one (batch, 256-prediction chunk).
// GT tile (8 KB) staged to LDS via gfx1250 async global->LDS (ASYNCcnt path),
// converted once to packed corner float4 + area, then a 256x512 IoU scan with
// 2 LDS issues (b128 + b32 broadcast) + ~20 VALU ops per pair.
// ---------------------------------------------------------------------------
__global__ __launch_bounds__(BLK) void yolo_main_kernel(
    const float4* __restrict__ yhat,    // (B, N) xywh
    const float4* __restrict__ ygt,     // (B, M) xywh
    const float*  __restrict__ scores,  // (B, N)
    const int*    __restrict__ minS,    // scalar
    float* __restrict__ ws)             // (B*32, 4) block partials
{
    __shared__ float4 rawGT[N_GT];      // 8 KB  (raw xywh, kept for MSE gather)
    __shared__ float4 gcorn[N_GT];      // 8 KB  (x0, y0, x1, y1)
    __shared__ float  gar[N_GT];        // 2 KB  (area)
    __shared__ float  redbuf[NWAVES * 4];

    const int b = blockIdx.y;
    const int t = threadIdx.x;
    const int n = blockIdx.x * BLK + t;

    // ---- stage GT boxes for this batch into LDS asynchronously ----
    const float4* gsrc = ygt + (size_t)b * N_GT;
    #pragma unroll
    for (int m = t; m < N_GT; m += BLK) {
        uint32_t lds_addr = (uint32_t)(size_t)(&rawGT[m]);   // low 32 bits = LDS offset
        uint64_t gaddr    = (uint64_t)(size_t)(gsrc + m);
        asm volatile("global_load_async_to_lds_b128 %0, %1, off"
                     :: "v"(lds_addr), "v"(gaddr) : "memory");
    }
    asm volatile("s_wait_asynccnt 0" ::: "memory");
    __syncthreads();

    // ---- convert GT to packed corner form + area (same formulas as reference) ----
    #pragma unroll
    for (int m = t; m < N_GT; m += BLK) {
        float4 g = rawGT[m];
        float x0 = g.x - g.z * 0.5f;    // *0.5 exact -> contraction-safe
        float y0 = g.y - g.w * 0.5f;
        float x1 = g.x + g.z * 0.5f;
        float y1 = g.y + g.w * 0.5f;
        gcorn[m] = make_float4(x0, y0, x1, y1);
        gar[m]   = (x1 - x0) * (y1 - y0);
    }
    __syncthreads();

    // ---- my prediction box ----
    float4 p = yhat[(size_t)b * N_PRED + n];
    float px0 = p.x - p.z * 0.5f;
    float py0 = p.y - p.w * 0.5f;
    float px1 = p.x + p.z * 0.5f;
    float py1 = p.y + p.w * 0.5f;
    float parea = (px1 - px0) * (py1 - py0);

    // ---- max / argmax IoU over all 512 GTs (strict > keeps first occurrence,
    //      matching jnp.argmax; per-pair IEEE division keeps iou bit-exact) ----
    float best = -1.0f;
    int   bidx = 0;
    #pragma unroll 8
    for (int m = 0; m < N_GT; ++m) {
        float4 c = gcorn[m];            // one ds_load_b128 (broadcast)
        float ga = gar[m];              // one ds_load_b32  (broadcast)
        float ltx = fmaxf(px0, c.x);
        float lty = fmaxf(py0, c.y);
        float rbx = fminf(px1, c.z);
        float rby = fminf(py1, c.w);
        float w = fmaxf(rbx - ltx, 0.0f);
        float h = fmaxf(rby - lty, 0.0f);
        float inter = w * h;
        float uni   = parea + ga - inter;
        float iou   = inter / uni;
        if (iou > best) { best = iou; bidx = m; }
    }

    // ---- per-prediction loss terms ----
    float ms  = (float)minS[0];
    float sc  = scores[(size_t)b * N_PRED + n];
    bool ixk     = sc > ms;
    bool matched = ixk && (best > IOU_THR);

    float4 g = rawGT[bidx];                 // matched GT in raw xywh form
    float dx = p.x - g.x, dy = p.y - g.y, dz = p.z - g.z, dw = p.w - g.w;
    float sq = dx * dx + dy * dy + dz * dz + dw * dw;

    float zlab = matched ? 1.0f : 0.0f;
    float bce  = fmaxf(sc, 0.0f) - sc * zlab + log1pf(expf(-fabsf(sc)));

    float v_sq  = matched ? sq  : 0.0f;
    float v_nm  = zlab;
    float v_bce = ixk ? bce  : 0.0f;
    float v_nf  = ixk ? 1.0f : 0.0f;

    // ---- deterministic block reduction: wave32 shfl tree, then cross-wave ----
    #pragma unroll
    for (int off = 16; off > 0; off >>= 1) {
        v_sq  += __shfl_down(v_sq,  off, 32);
        v_nm  += __shfl_down(v_nm,  off, 32);
        v_bce += __shfl_down(v_bce, off, 32);
        v_nf  += __shfl_down(v_nf,  off, 32);
    }
    int wave = t >> 5, lane = t & 31;
    if (lane == 0) {
        redbuf[wave * 4 + 0] = v_sq;
        redbuf[wave * 4 + 1] = v_nm;
        redbuf[wave * 4 + 2] = v_bce;
        redbuf[wave * 4 + 3] = v_nf;
    }
    __syncthreads();
    if (t == 0) {
        float s0 = 0.f, s1 = 0.f, s2 = 0.f, s3 = 0.f;
        #pragma unroll
        for (int wv = 0; wv < NWAVES; ++wv) {
            s0 += redbuf[wv * 4 + 0];
            s1 += redbuf[wv * 4 + 1];
            s2 += redbuf[wv * 4 + 2];
            s3 += redbuf[wv * 4 + 3];
        }
        int pb = b * BLOCKS_PER_BATCH + blockIdx.x;
        ws[pb * 4 + 0] = s0;   // sum sq_err * matched
        ws[pb * 4 + 1] = s1;   // n_matched
        ws[pb * 4 + 2] = s2;   // sum bce * ix
        ws[pb * 4 + 3] = s3;   // n_filt
    }
}

// ---------------------------------------------------------------------------
// Final kernel: 16 wave32s; wave b reduces batch b's 32 block-partials,
// applies the per-batch loss formulas, thread 0 writes total / B.
// ---------------------------------------------------------------------------
__global__ __launch_bounds__(512) void yolo_final_kernel(
    const float* __restrict__ ws, float* __restrict__ out)
{
    __shared__ float acc[NBATCH];
    int t = threadIdx.x;
    int bb = t >> 5, lane = t & 31;

    float s_sq  = ws[(bb * 32 + lane) * 4 + 0];
    float s_nm  = ws[(bb * 32 + lane) * 4 + 1];
    float s_bce = ws[(bb * 32 + lane) * 4 + 2];
    float s_nf  = ws[(bb * 32 + lane) * 4 + 3];
    #pragma unroll
    for (int off = 16; off > 0; off >>= 1) {
        s_sq  += __shfl_down(s_sq,  off, 32);
        s_nm  += __shfl_down(s_nm,  off, 32);
        s_bce += __shfl_down(s_bce, off, 32);
        s_nf  += __shfl_down(s_nf,  off, 32);
    }
    if (lane == 0) {
        float loc = (s_nm > 0.0f) ? s_sq / (4.0f * fmaxf(s_nm, 1.0f)) : 0.0f;
        float obj = s_bce / fmaxf(s_nf, 1.0f);
        float pen = PEN_FAC * (s_nf - s_nm);
        acc[bb] = loc + obj + pen;
    }
    __syncthreads();
    if (t == 0) {
        float tot = 0.0f;
        #pragma unroll
        for (int i = 0; i < NBATCH; ++i) tot += acc[i];
        out[0] = tot / (float)NBATCH;
    }
}

extern "C" void kernel_launch(void* const* d_in, const int* in_sizes, int n_in,
                              void* d_out, int out_size, void* d_ws, size_t ws_size,
                              hipStream_t stream) {
    const float4* yhat   = (const float4*)d_in[0];   // (16, 8192, 4) f32
    const float4* ygt    = (const float4*)d_in[1];   // (16, 512, 4)  f32
    const float*  scores = (const float*)d_in[2];    // (16, 8192)    f32
    const int*    minS   = (const int*)d_in[3];      // scalar int 0
    float* ws  = (float*)d_ws;                       // 512 blocks * 4 floats
    float* out = (float*)d_out;

    dim3 grid(BLOCKS_PER_BATCH, NBATCH);
    yolo_main_kernel<<<grid, BLK, 0, stream>>>(yhat, ygt, scores, minS, ws);
    yolo_final_kernel<<<1, 512, 0, stream>>>(ws, out);
}